// GraphableDecodeStep_35665408426432
// MI455X (gfx1250) — compile-verified
//
#include <hip/hip_runtime.h>
#include <math.h>

#define NLAYERS 12
#define DMODEL  1024
#define NHEADS  16
#define NKV     4
#define HEADD   64
#define FFDIM   4096
#define VOCAB   32000
#define SMAX    2048

typedef __attribute__((ext_vector_type(2))) float v2f;
typedef __attribute__((ext_vector_type(8))) float v8f;
typedef __attribute__((ext_vector_type(4))) int   v4i;

// ---------------------------------------------------------------- embed
__global__ void embed_kernel(const int* __restrict__ ids,
                             const float* __restrict__ embed,
                             float* __restrict__ h) {
  const int t = ids[0];
  for (int i = threadIdx.x; i < DMODEL; i += blockDim.x)
    h[i] = embed[(size_t)t * DMODEL + i];
}

// ---------------------------------------------------------------- rmsnorm
__global__ void rmsnorm_kernel(const float* __restrict__ x,
                               const float* __restrict__ w,
                               float* __restrict__ out, int n) {
  __shared__ float buf[256];
  float s = 0.f;
  for (int i = threadIdx.x; i < n; i += 256) { float v = x[i]; s += v * v; }
  buf[threadIdx.x] = s;
  __syncthreads();
  for (int off = 128; off > 0; off >>= 1) {
    if (threadIdx.x < off) buf[threadIdx.x] += buf[threadIdx.x + off];
    __syncthreads();
  }
  const float inv = rsqrtf(buf[0] / (float)n + 1e-6f);
  for (int i = threadIdx.x; i < n; i += 256) out[i] = x[i] * inv * w[i];
}

// ---------------------------------------------------------------- GEMV via WMMA f32 16x16x4
// y[n] = sum_k x[k] * W[k*N + n]  (+ res[n])
// One wave -> 16 output columns. A-tile: row M=0 = x[k0..k0+3], rest 0.
// B-tile: 4x16 chunk of W streamed with NT hint. Accumulate C over K in steps of 4.
__global__ void gemv_wmma_kernel(const float* __restrict__ x,
                                 const float* __restrict__ W,
                                 const float* __restrict__ res,
                                 float* __restrict__ y,
                                 int K, int N) {
  __shared__ float xs[FFDIM];  // K <= 4096

  // Stage x into LDS via the CDNA5 async-DMA path (ASYNCcnt-tracked),
  // 16 B per lane per op; falls back to plain copy if builtins absent.
#if defined(__AMDGCN__) && __has_builtin(__builtin_amdgcn_global_load_async_to_lds_b128) && \
    __has_builtin(__builtin_amdgcn_s_wait_asynccnt)
  for (int i = threadIdx.x * 4; i < K; i += blockDim.x * 4) {
    __builtin_amdgcn_global_load_async_to_lds_b128(
        (v4i*)(x + i),
        (v4i*)(xs + i),
        /*imm offset=*/0, /*cpol=*/0);
  }
  __builtin_amdgcn_s_wait_asynccnt(0);
#else
  for (int i = threadIdx.x * 4; i < K; i += blockDim.x * 4) {
    xs[i + 0] = x[i + 0];
    xs[i + 1] = x[i + 1];
    xs[i + 2] = x[i + 2];
    xs[i + 3] = x[i + 3];
  }
#endif
  __syncthreads();

  const int lane = threadIdx.x & 31;
  const int wid  = threadIdx.x >> 5;
  const int n0   = (blockIdx.x * (blockDim.x >> 5) + wid) * 16;  // grids sized exactly; no guard -> EXEC stays all-ones
  const int half = lane >> 4;   // 0: lanes 0-15 (K slot 0/1), 1: lanes 16-31 (K slot 2/3)
  const int l16  = lane & 15;

  const float* Wp = W + (size_t)n0 + l16;
  v8f acc = {};
#pragma unroll 8
  for (int k0 = 0; k0 < K; k0 += 4) {
    v2f a, b;
    const int kh = k0 + 2 * half;
    a.x = (l16 == 0) ? xs[kh]     : 0.f;   // A: 16x4, only M=0 row non-zero
    a.y = (l16 == 0) ? xs[kh + 1] : 0.f;
    // weights are a one-shot stream: non-temporal so they don't evict
    // KV-cache/activations from the 192MB L2
    b.x = __builtin_nontemporal_load(Wp + (size_t)kh * N);
    b.y = __builtin_nontemporal_load(Wp + (size_t)(kh + 1) * N);
    acc = __builtin_amdgcn_wmma_f32_16x16x4_f32(
        /*neg_a=*/false, a, /*neg_b=*/false, b,
        /*c_mod=*/(short)0, acc, /*reuse_a=*/false, /*reuse_b=*/false);
  }
  // D row M=0 lives in VGPR0 (acc[0]) of lanes 0-15, N = n0 + lane
  if (half == 0) {
    float r = acc[0];
    if (res) r += res[n0 + l16];
    y[n0 + l16] = r;
  }
}

// ---------------------------------------------------------------- RoPE (q: 16 heads, k: 4 heads)
__global__ void rope_kernel(float* __restrict__ q, float* __restrict__ k,
                            const int* __restrict__ cache_len) {
  const int pos = cache_len[0];
  const int t = blockIdx.x * blockDim.x + threadIdx.x;
  const int total = (NHEADS + NKV) * (HEADD / 2);
  if (t >= total) return;
  const int head = t >> 5;          // HEADD/2 == 32 pairs per head
  const int j    = t & 31;
  float* base = (head < NHEADS) ? (q + head * HEADD) : (k + (head - NHEADS) * HEADD);
  const float invf = __powf(10000.0f, -(float)(2 * j) / (float)HEADD);
  const float ang = (float)pos * invf;
  const float c = __cosf(ang), s = __sinf(ang);
  const float a = base[j], b = base[j + 32];
  base[j]      = a * c - b * s;     // x*cos + rotate_half(x)*sin
  base[j + 32] = b * c + a * s;
}

// ---------------------------------------------------------------- attention (one block per kv head)
__global__ void attention_kernel(const float* __restrict__ q,
                                 const float* __restrict__ knew,
                                 const float* __restrict__ vnew,
                                 const float* __restrict__ kcache,
                                 const float* __restrict__ vcache,
                                 const int* __restrict__ cache_len,
                                 float* __restrict__ out) {
  __shared__ float sc[4][SMAX];   // 32 KB of the 320 KB/WGP LDS
  __shared__ float red[4][256];
  const int g   = blockIdx.x;     // kv head
  const int tid = threadIdx.x;    // 256 threads = 8 wave32
  const int pos = cache_len[0];
  const float scale = 0.125f;     // 1/sqrt(64)
  const float* Kg = kcache + (size_t)g * SMAX * HEADD;
  const float* Vg = vcache + (size_t)g * SMAX * HEADD;

  // scores for the 4 query heads sharing this kv head
  for (int s = tid; s <= pos; s += 256) {
    const float* kr = (s == pos) ? (knew + g * HEADD) : (Kg + (size_t)s * HEADD);
    for (int qh = 0; qh < 4; ++qh) {
      const float* qr = q + (g * 4 + qh) * HEADD;
      float d = 0.f;
      for (int i = 0; i < HEADD; ++i) d += qr[i] * kr[i];
      sc[qh][s] = d * scale;
    }
  }
  __syncthreads();

  // softmax: block max
  float lmax[4] = {-1e30f, -1e30f, -1e30f, -1e30f};
  for (int s = tid; s <= pos; s += 256)
    for (int qh = 0; qh < 4; ++qh) lmax[qh] = fmaxf(lmax[qh], sc[qh][s]);
  for (int qh = 0; qh < 4; ++qh) red[qh][tid] = lmax[qh];
  __syncthreads();
  for (int off = 128; off > 0; off >>= 1) {
    if (tid < off)
      for (int qh = 0; qh < 4; ++qh)
        red[qh][tid] = fmaxf(red[qh][tid], red[qh][tid + off]);
    __syncthreads();
  }
  float gmax[4];
  for (int qh = 0; qh < 4; ++qh) gmax[qh] = red[qh][0];
  __syncthreads();

  // exp + block sum
  float lsum[4] = {0.f, 0.f, 0.f, 0.f};
  for (int s = tid; s <= pos; s += 256)
    for (int qh = 0; qh < 4; ++qh) {
      float e = __expf(sc[qh][s] - gmax[qh]);
      sc[qh][s] = e;
      lsum[qh] += e;
    }
  for (int qh = 0; qh < 4; ++qh) red[qh][tid] = lsum[qh];
  __syncthreads();
  for (int off = 128; off > 0; off >>= 1) {
    if (tid < off)
      for (int qh = 0; qh < 4; ++qh) red[qh][tid] += red[qh][tid + off];
    __syncthreads();
  }
  float ginv[4];
  for (int qh = 0; qh < 4; ++qh) ginv[qh] = 1.f / red[qh][0];
  __syncthreads();

  // attn @ V : thread -> (qh, d); coalesced 128B V reads per wave per position
  const int qh = tid >> 6;
  const int dd = tid & 63;
  float acc = 0.f;
  for (int s = 0; s < pos; ++s) acc += sc[qh][s] * Vg[(size_t)s * HEADD + dd];
  acc += sc[qh][pos] * vnew[g * HEADD + dd];
  out[(g * 4 + qh) * HEADD + dd] = acc * ginv[qh];
}

// ---------------------------------------------------------------- SwiGLU elementwise
__global__ void silu_mul_kernel(const float* __restrict__ g,
                                const float* __restrict__ u,
                                float* __restrict__ out, int n) {
  const int i = blockIdx.x * blockDim.x + threadIdx.x;
  if (i < n) {
    const float x = g[i];
    out[i] = (x / (1.f + __expf(-x))) * u[i];
  }
}

// ---------------------------------------------------------------- host orchestration
extern "C" void kernel_launch(void* const* d_in, const int* in_sizes, int n_in,
                              void* d_out, int out_size, void* d_ws, size_t ws_size,
                              hipStream_t stream) {
  const int*   ids    = (const int*)  d_in[0];
  const float* embedw = (const float*)d_in[1];
  const float* w_inln = (const float*)d_in[2];
  const float* wq     = (const float*)d_in[3];
  const float* wk     = (const float*)d_in[4];
  const float* wv     = (const float*)d_in[5];
  const float* wo     = (const float*)d_in[6];
  const float* wpost  = (const float*)d_in[7];
  const float* wgate  = (const float*)d_in[8];
  const float* wup    = (const float*)d_in[9];
  const float* wdown  = (const float*)d_in[10];
  const float* finln  = (const float*)d_in[11];
  const float* lmh    = (const float*)d_in[12];
  const float* kc     = (const float*)d_in[13];
  const float* vc     = (const float*)d_in[14];
  const int*   clen   = (const int*)  d_in[15];
  float* out = (float*)d_out;
  float* ws  = (float*)d_ws;

  float* h  = ws;            // 1024
  float* xb = ws + 1024;     // 1024
  float* qb = ws + 2048;     // 1024
  float* kn = ws + 3072;     // 256
  float* vn = ws + 3328;     // 256
  float* at = ws + 3584;     // 1024
  float* x2 = ws + 4608;     // 1024
  float* gb = ws + 5632;     // 4096
  float* ub = ws + 9728;     // 4096
  float* ff = ws + 13824;    // 4096

  embed_kernel<<<1, 256, 0, stream>>>(ids, embedw, h);

  for (int l = 0; l < NLAYERS; ++l) {
    rmsnorm_kernel<<<1, 256, 0, stream>>>(h, w_inln + l * DMODEL, xb, DMODEL);

    gemv_wmma_kernel<<<8, 256, 0, stream>>>(
        xb, wq + (size_t)l * DMODEL * (NHEADS * HEADD), nullptr, qb, DMODEL, NHEADS * HEADD);
    gemv_wmma_kernel<<<2, 256, 0, stream>>>(
        xb, wk + (size_t)l * DMODEL * (NKV * HEADD), nullptr, kn, DMODEL, NKV * HEADD);
    gemv_wmma_kernel<<<2, 256, 0, stream>>>(
        xb, wv + (size_t)l * DMODEL * (NKV * HEADD), nullptr, vn, DMODEL, NKV * HEADD);

    rope_kernel<<<3, 256, 0, stream>>>(qb, kn, clen);

    attention_kernel<<<NKV, 256, 0, stream>>>(
        qb, kn, vn,
        kc + (size_t)l * NKV * SMAX * HEADD,
        vc + (size_t)l * NKV * SMAX * HEADD,
        clen, at);

    gemv_wmma_kernel<<<8, 256, 0, stream>>>(
        at, wo + (size_t)l * (NHEADS * HEADD) * DMODEL, h, h, NHEADS * HEADD, DMODEL);

    rmsnorm_kernel<<<1, 256, 0, stream>>>(h, wpost + l * DMODEL, x2, DMODEL);

    gemv_wmma_kernel<<<32, 256, 0, stream>>>(
        x2, wgate + (size_t)l * DMODEL * FFDIM, nullptr, gb, DMODEL, FFDIM);
    gemv_wmma_kernel<<<32, 256, 0, stream>>>(
        x2, wup + (size_t)l * DMODEL * FFDIM, nullptr, ub, DMODEL, FFDIM);
    silu_mul_kernel<<<FFDIM / 256, 256, 0, stream>>>(gb, ub, ff, FFDIM);

    gemv_wmma_kernel<<<8, 256, 0, stream>>>(
        ff, wdown + (size_t)l * FFDIM * DMODEL, h, h, FFDIM, DMODEL);
  }

  rmsnorm_kernel<<<1, 256, 0, stream>>>(h, finln, xb, DMODEL);
  // 32000 cols / 128 cols-per-block = 250 blocks, exact
  gemv_wmma_kernel<<<250, 256, 0, stream>>>(xb, lmh, nullptr, out, DMODEL, VOCAB);
}